// EdgeConv2d_34368328302639
// MI455X (gfx1250) — compile-verified
//
#include <hip/hip_runtime.h>

typedef float v2f __attribute__((ext_vector_type(2)));
typedef float v4f __attribute__((ext_vector_type(4)));
typedef float v8f __attribute__((ext_vector_type(8)));

constexpr int B_  = 8;
constexpr int C_  = 128;   // input channels
constexpr int N_  = 4096;  // nodes
constexpr int K_  = 16;    // neighbors
constexpr int CO_ = 256;   // output channels

// ---------------------------------------------------------------------------
// Phase 0a: k-pair-interleaved weights.
//   Wc2[(k2*CO + o)*2 + e] = W[o][2k2+e] - W[o][128+2k2+e]
//   Wn2[(k2*CO + o)*2 + e] = W[o][128+2k2+e]
// One b64 load then feeds a whole WMMA B operand.
// ---------------------------------------------------------------------------
__global__ void prep_w(const float* __restrict__ W,
                       float* __restrict__ Wc2, float* __restrict__ Wn2) {
  const int k2 = blockIdx.x;   // 0..63
  const int o  = threadIdx.x;  // 0..255
  const float w1a = W[o * (2 * C_) + 2 * k2];
  const float w1b = W[o * (2 * C_) + 2 * k2 + 1];
  const float w2a = W[o * (2 * C_) + C_ + 2 * k2];
  const float w2b = W[o * (2 * C_) + C_ + 2 * k2 + 1];
  v2f c; c.x = w1a - w2a; c.y = w1b - w2b;
  v2f n; n.x = w2a;       n.y = w2b;
  *(v2f*)&Wc2[((size_t)k2 * CO_ + o) * 2] = c;
  *(v2f*)&Wn2[((size_t)k2 * CO_ + o) * 2] = n;
}

// ---------------------------------------------------------------------------
// Phase 0b: k-pair-interleaved x copy: xp[((b*64 + c2)*N + n)*2 + e] = x[b][2c2+e][n]
// ---------------------------------------------------------------------------
__global__ void __launch_bounds__(256) prep_x(const float* __restrict__ x,
                                              float* __restrict__ xp) {
  const int blk = blockIdx.x;
  const int nb  = blk & 15;          // 16 n-chunks of 256
  const int bc2 = blk >> 4;          // b*64 + c2
  const int b   = bc2 >> 6;
  const int c2  = bc2 & 63;
  const int n   = nb * 256 + threadIdx.x;
  v2f a;
  a.x = x[((size_t)b * C_ + 2 * c2) * N_ + n];
  a.y = x[((size_t)b * C_ + 2 * c2 + 1) * N_ + n];
  *(v2f*)&xp[((size_t)bc2 * N_ + n) * 2] = a;
}

// ---------------------------------------------------------------------------
// Phase 1: per batch,  U^T(4096x256) = x^T(4096x128) @ Wc(128x256)
//                      V^T(4096x256) = x^T(4096x128) @ Wn(128x256)
// One wave per (16-node tile, 64-output-channel group): 4 U + 4 V tiles
// sharing A loads. f32 WMMA 16x16x4; K-loop unrolled x2 with explicit
// ping-pong operand buffers (no register-rotation moves), all operands b64.
//   A[m][k]: m = lane&15, k = 2*(lane>>4) + vgpr
//   B[k][n]: n = lane&15, k = 2*(lane>>4) + vgpr
//   D[m][n]: n = lane&15, m = vgpr + 8*(lane>>4)
// ---------------------------------------------------------------------------
__global__ void __launch_bounds__(256) gemm_uv(const float* __restrict__ xp,
                                               const float* __restrict__ Wc2,
                                               const float* __restrict__ Wn2,
                                               float* __restrict__ U,
                                               float* __restrict__ V) {
  const int wid  = blockIdx.x * 8 + (threadIdx.x >> 5);
  const int lane = threadIdx.x & 31;
  const int row  = lane & 15;
  const int grp  = lane >> 4;

  const int b      = wid >> 10;        // 256 node tiles * 4 o-groups per batch
  const int rem    = wid & 1023;
  const int mt     = rem >> 2;         // node tile 0..255
  const int o_base = (rem & 3) * 64;   // output-channel group base

  const float* xpb   = xp + (size_t)b * (C_ / 2) * N_ * 2;
  const int    node0 = mt * 16;

  v8f accU[4] = {};
  v8f accV[4] = {};

  auto loadA = [&](int k0) -> v2f {
    const int k2 = (k0 >> 1) + grp;
    return *(const v2f*)(xpb + ((size_t)k2 * N_ + node0 + row) * 2);
  };
  auto loadB = [&](const float* Wt, int k0, int ot) -> v2f {
    const int k2 = (k0 >> 1) + grp;
    return *(const v2f*)(Wt + ((size_t)k2 * CO_ + o_base + ot * 16 + row) * 2);
  };

  // Stage 0 preload.
  v2f a0 = loadA(0);
  v2f bu0[4], bv0[4], bu1[4], bv1[4];
#pragma unroll
  for (int ot = 0; ot < 4; ++ot) {
    bu0[ot] = loadB(Wc2, 0, ot);
    bv0[ot] = loadB(Wn2, 0, ot);
  }

  for (int k0 = 0; k0 < C_; k0 += 8) {
    // Prefetch stage 1 (k0+4), then consume stage 0.
    v2f a1 = loadA(k0 + 4);
#pragma unroll
    for (int ot = 0; ot < 4; ++ot) {
      bu1[ot] = loadB(Wc2, k0 + 4, ot);
      bv1[ot] = loadB(Wn2, k0 + 4, ot);
    }
#pragma unroll
    for (int ot = 0; ot < 4; ++ot) {
      accU[ot] = __builtin_amdgcn_wmma_f32_16x16x4_f32(
          false, a0, false, bu0[ot], (short)0, accU[ot], false, false);
      accV[ot] = __builtin_amdgcn_wmma_f32_16x16x4_f32(
          false, a0, false, bv0[ot], (short)0, accV[ot], false, false);
    }
    // Prefetch stage 0 of next iteration (k0+8, wraps on last iter), consume stage 1.
    const int kn = (k0 + 8) & (C_ - 1);
    a0 = loadA(kn);
#pragma unroll
    for (int ot = 0; ot < 4; ++ot) {
      bu0[ot] = loadB(Wc2, kn, ot);
      bv0[ot] = loadB(Wn2, kn, ot);
    }
#pragma unroll
    for (int ot = 0; ot < 4; ++ot) {
      accU[ot] = __builtin_amdgcn_wmma_f32_16x16x4_f32(
          false, a1, false, bu1[ot], (short)0, accU[ot], false, false);
      accV[ot] = __builtin_amdgcn_wmma_f32_16x16x4_f32(
          false, a1, false, bv1[ot], (short)0, accV[ot], false, false);
    }
  }

  // Store node-major: U[b][node][o] (contiguous 256 floats per node).
#pragma unroll
  for (int ot = 0; ot < 4; ++ot) {
    const int o0 = o_base + ot * 16;
#pragma unroll
    for (int r = 0; r < 8; ++r) {
      const int    node = node0 + r + grp * 8;
      const size_t off  = ((size_t)b * N_ + node) * CO_ + o0 + row;
      U[off] = accU[ot][r];
      V[off] = accV[ot][r];
    }
  }
}

// ---------------------------------------------------------------------------
// Phase 2: out[b][o][n] = relu( max_k (U[b][i_k][o] + V[b][j_k][o]) + bias[o] )
// 16 nodes per block. Each thread owns a channel QUAD (b128 gathers; a
// 32-lane group reads 512B contiguous per row) over 4 nodes. LDS transpose
// with stride-260 padding (conflict-free for quad stores and transposed quad
// reads); b128 output writes coalesced over node quads.
// ---------------------------------------------------------------------------
__global__ void __launch_bounds__(256) gather_max(const long long* __restrict__ ei,
                                                  const float* __restrict__ U,
                                                  const float* __restrict__ V,
                                                  const float* __restrict__ bias,
                                                  float* __restrict__ out) {
  __shared__ int   si[16][16];
  __shared__ int   sj[16][16];
  __shared__ float so[16][260];

  const int b   = blockIdx.x >> 8;          // 256 node tiles per batch
  const int n0  = (blockIdx.x & 255) * 16;
  const int tid = threadIdx.x;

  {
    const int    nn   = tid >> 4, k = tid & 15;
    const size_t base = ((size_t)b * N_ + (n0 + nn)) * K_ + k;
    sj[nn][k] = (int)ei[base];                          // edge_index[0]: neighbor (x_j)
    si[nn][k] = (int)ei[(size_t)B_ * N_ * K_ + base];   // edge_index[1]: center  (x_i)
  }
  __syncthreads();

  const int c4 = tid & 63;    // channel quad: channels 4c4 .. 4c4+3
  const int h  = tid >> 6;    // node quarter: nodes h*4 .. h*4+3
  const v4f bo = *(const v4f*)&bias[4 * c4];
  const float* Ub = U + (size_t)b * N_ * CO_;
  const float* Vb = V + (size_t)b * N_ * CO_;

#pragma unroll 1
  for (int nn = h * 4; nn < h * 4 + 4; ++nn) {
    v4f m; m.x = m.y = m.z = m.w = -3.4e38f;
#pragma unroll
    for (int k = 0; k < K_; ++k) {
      const v4f u = *(const v4f*)&Ub[(size_t)si[nn][k] * CO_ + 4 * c4];
      const v4f v = *(const v4f*)&Vb[(size_t)sj[nn][k] * CO_ + 4 * c4];
      m.x = fmaxf(m.x, u.x + v.x);
      m.y = fmaxf(m.y, u.y + v.y);
      m.z = fmaxf(m.z, u.z + v.z);
      m.w = fmaxf(m.w, u.w + v.w);
    }
    v4f r;
    r.x = fmaxf(m.x + bo.x, 0.0f);
    r.y = fmaxf(m.y + bo.y, 0.0f);
    r.z = fmaxf(m.z + bo.z, 0.0f);
    r.w = fmaxf(m.w + bo.w, 0.0f);
    *(v4f*)&so[nn][4 * c4] = r;
  }
  __syncthreads();

  // Transposed writes: thread -> (node quad nq, channel lane oo); b128 stores,
  // 4 consecutive lanes cover 64B of contiguous n.
  const int nq = tid & 3;     // node quad: nodes 4nq .. 4nq+3
  const int oo = tid >> 2;    // 0..63
#pragma unroll
  for (int og = 0; og < 4; ++og) {
    const int oc = og * 64 + oo;
    v4f w;
    w.x = so[4 * nq + 0][oc];
    w.y = so[4 * nq + 1][oc];
    w.z = so[4 * nq + 2][oc];
    w.w = so[4 * nq + 3][oc];
    *(v4f*)&out[((size_t)b * CO_ + oc) * N_ + n0 + 4 * nq] = w;
  }
}

// ---------------------------------------------------------------------------
extern "C" void kernel_launch(void* const* d_in, const int* in_sizes, int n_in,
                              void* d_out, int out_size, void* d_ws, size_t ws_size,
                              hipStream_t stream) {
  const float*     x    = (const float*)d_in[0];
  const long long* ei   = (const long long*)d_in[1];
  const float*     W    = (const float*)d_in[2];
  const float*     bias = (const float*)d_in[3];

  float* ws  = (float*)d_ws;
  float* Wc2 = ws;                                     //  32768 floats
  float* Wn2 = ws + 128 * 256;                         //  32768 floats
  float* xp  = ws + 2 * 128 * 256;                     //  4M floats (16 MB)
  float* U   = xp + (size_t)B_ * (C_ / 2) * N_ * 2;    //  8M floats (32 MB)
  float* V   = U + (size_t)B_ * N_ * CO_;              //  8M floats (32 MB)

  prep_w<<<64, 256, 0, stream>>>(W, Wc2, Wn2);
  prep_x<<<8192, 256, 0, stream>>>(x, xp);
  gemm_uv<<<1024, 256, 0, stream>>>(xp, Wc2, Wn2, U, V);
  gather_max<<<2048, 256, 0, stream>>>(ei, U, V, bias, (float*)d_out);
}